// HeteroSAGELayer_24180665876652
// MI455X (gfx1250) — compile-verified
//
#include <hip/hip_runtime.h>
#include <hip/hip_bf16.h>

typedef __attribute__((ext_vector_type(2))) float v2f;
typedef __attribute__((ext_vector_type(8))) float v8f;

#define DDIM 128

// ---------------------------------------------------------------------------
// Y[M x 128] = relu(X[M x 128] @ W^T), W row-major [128 x 128].
// Block = 256 threads = 8 waves; wave w computes cols [16w,16w+16) of a
// 16-row tile. K loop of 32 chained v_wmma_f32_16x16x4_f32.
// ---------------------------------------------------------------------------
__global__ __launch_bounds__(256)
void gemm_relu_xwt(const float* __restrict__ X, const float* __restrict__ W,
                   float* __restrict__ Y) {
  const int tid   = threadIdx.x;
  const int wave  = tid >> 5;        // 0..7 -> N tile
  const int lane  = tid & 31;
  const int lrow  = lane & 15;
  const int khalf = (lane >> 4) * 2; // 0 or 2
  const int m0 = blockIdx.x * 16;
  const int n0 = wave * 16;

  const float* Arow = X + (size_t)(m0 + lrow) * DDIM; // A row for this lane
  const float* Brow = W + (size_t)(n0 + lrow) * DDIM; // B=W^T -> W row n

  v8f acc = {};
#pragma unroll
  for (int k = 0; k < DDIM; k += 4) {
    v2f a = *(const v2f*)(Arow + k + khalf);
    v2f b = *(const v2f*)(Brow + k + khalf);
    acc = __builtin_amdgcn_wmma_f32_16x16x4_f32(false, a, false, b,
                                                (short)0, acc, false, false);
  }
  const int mhi = (lane >> 4) * 8;
#pragma unroll
  for (int v = 0; v < 8; ++v) {
    float r = acc[v];
    r = r > 0.f ? r : 0.f;
    Y[(size_t)(m0 + mhi + v) * DDIM + n0 + lrow] = r;
  }
}

// ---------------------------------------------------------------------------
// Scatter-add: one 32-lane group per edge; each lane moves 4 floats.
// ---------------------------------------------------------------------------
__global__ __launch_bounds__(256)
void scatter_add(const float* __restrict__ Y, const int* __restrict__ src,
                 const int* __restrict__ dst, float* __restrict__ S,
                 float* __restrict__ cnt, int E) {
  int e = blockIdx.x * 8 + (threadIdx.x >> 5);
  if (e >= E) return;
  int lane = threadIdx.x & 31;
  int s = src[e];
  int d = dst[e];
  const float4 v = *(const float4*)(Y + (size_t)s * DDIM + lane * 4);
  float* p = S + (size_t)d * DDIM + lane * 4;
  atomicAdd(p + 0, v.x);
  atomicAdd(p + 1, v.y);
  atomicAdd(p + 2, v.z);
  atomicAdd(p + 3, v.w);
  if (lane == 0) atomicAdd(cnt + d, 1.0f);
}

// agg_drug = 0.5*(S_dd/max(c_dd,1) + S_pd/max(c_pd,1)) -> in place into S_dd
__global__ __launch_bounds__(256)
void combine_drug(float* __restrict__ S_dd, const float* __restrict__ c_dd,
                  const float* __restrict__ S_pd, const float* __restrict__ c_pd,
                  int N) {
  size_t i = (size_t)blockIdx.x * blockDim.x + threadIdx.x;
  if (i >= (size_t)N * DDIM) return;
  int row = (int)(i >> 7);
  float cd = c_dd[row]; cd = cd > 1.f ? cd : 1.f;
  float cp = c_pd[row]; cp = cp > 1.f ? cp : 1.f;
  S_dd[i] = 0.5f * (S_dd[i] / cd + S_pd[i] / cp);
}

// S /= max(c,1)  (in place)
__global__ __launch_bounds__(256)
void normalize_rows(float* __restrict__ S, const float* __restrict__ c, int N) {
  size_t i = (size_t)blockIdx.x * blockDim.x + threadIdx.x;
  if (i >= (size_t)N * DDIM) return;
  int row = (int)(i >> 7);
  float cc = c[row]; cc = cc > 1.f ? cc : 1.f;
  S[i] = S[i] / cc;
}

__device__ __forceinline__ float wave_reduce_sum(float v) {
#pragma unroll
  for (int m = 16; m >= 1; m >>= 1) v += __shfl_xor(v, m, 32);
  return v;
}

// ---------------------------------------------------------------------------
// Node update: h = relu([x,agg] @ W^T + b); out = LN(h + x, g, be).
// W row-major [128 x 256]. Block = 256 threads, 16 nodes per block.
// GEMM (K=256, fused bias/relu/residual) -> LDS -> wave32 layernorm.
// ---------------------------------------------------------------------------
__global__ __launch_bounds__(256)
void node_update(const float* __restrict__ X, const float* __restrict__ AGG,
                 const float* __restrict__ W, const float* __restrict__ B,
                 const float* __restrict__ G, const float* __restrict__ BE,
                 float* __restrict__ OUT, size_t out_row_base) {
  __shared__ float tile[16 * 132];  // 132-float row pitch: conflict-free writes

  const int tid   = threadIdx.x;
  const int wave  = tid >> 5;
  const int lane  = tid & 31;
  const int lrow  = lane & 15;
  const int khalf = (lane >> 4) * 2;
  const int m0 = blockIdx.x * 16;
  const int n0 = wave * 16;

  const float* Xrow = X   + (size_t)(m0 + lrow) * DDIM;
  const float* Grow = AGG + (size_t)(m0 + lrow) * DDIM;
  const float* Wr   = W   + (size_t)(n0 + lrow) * (2 * DDIM);

  v8f acc = {};
#pragma unroll
  for (int k = 0; k < DDIM; k += 4) {   // first half of K: x_self
    v2f a = *(const v2f*)(Xrow + k + khalf);
    v2f b = *(const v2f*)(Wr + k + khalf);
    acc = __builtin_amdgcn_wmma_f32_16x16x4_f32(false, a, false, b,
                                                (short)0, acc, false, false);
  }
#pragma unroll
  for (int k = 0; k < DDIM; k += 4) {   // second half of K: agg
    v2f a = *(const v2f*)(Grow + k + khalf);
    v2f b = *(const v2f*)(Wr + DDIM + k + khalf);
    acc = __builtin_amdgcn_wmma_f32_16x16x4_f32(false, a, false, b,
                                                (short)0, acc, false, false);
  }

  const int mhi  = (lane >> 4) * 8;
  const float bj = B[n0 + lrow];
#pragma unroll
  for (int v = 0; v < 8; ++v) {
    int row = mhi + v;
    float h = acc[v] + bj;
    h = h > 0.f ? h : 0.f;                                  // relu
    h += X[(size_t)(m0 + row) * DDIM + n0 + lrow];          // residual
    tile[row * 132 + n0 + lrow] = h;
  }
  __syncthreads();

  // layernorm: wave w handles rows 2w, 2w+1; 4 elements per lane
#pragma unroll
  for (int rr = 0; rr < 2; ++rr) {
    int r = wave * 2 + rr;
    float4 vv = *(const float4*)&tile[r * 132 + lane * 4];
    float s = wave_reduce_sum(vv.x + vv.y + vv.z + vv.w);
    float mu = s * (1.0f / 128.0f);
    float d0 = vv.x - mu, d1 = vv.y - mu, d2 = vv.z - mu, d3 = vv.w - mu;
    float q = wave_reduce_sum(d0 * d0 + d1 * d1 + d2 * d2 + d3 * d3);
    float rstd = rsqrtf(q * (1.0f / 128.0f) + 1e-5f);
    float4 gg = *(const float4*)(G + lane * 4);
    float4 bb = *(const float4*)(BE + lane * 4);
    float4 o;
    o.x = d0 * rstd * gg.x + bb.x;
    o.y = d1 * rstd * gg.y + bb.y;
    o.z = d2 * rstd * gg.z + bb.z;
    o.w = d3 * rstd * gg.w + bb.w;
    *(float4*)(OUT + (out_row_base + (size_t)(m0 + r)) * DDIM + lane * 4) = o;
  }
}

// ---------------------------------------------------------------------------
extern "C" void kernel_launch(void* const* d_in, const int* in_sizes, int n_in,
                              void* d_out, int out_size, void* d_ws, size_t ws_size,
                              hipStream_t stream) {
  constexpr int N_DRUG = 100000;
  constexpr int N_PROT = 50000;

  const float* x_drug  = (const float*)d_in[0];
  const float* x_prot  = (const float*)d_in[1];
  const float* Wagg_dd = (const float*)d_in[2];
  const float* Wagg_dp = (const float*)d_in[3];
  const float* Wagg_pd = (const float*)d_in[4];
  const float* W_drug  = (const float*)d_in[5];
  const float* b_drug  = (const float*)d_in[6];
  const float* W_prot  = (const float*)d_in[7];
  const float* b_prot  = (const float*)d_in[8];
  const float* g_drug  = (const float*)d_in[9];
  const float* be_drug = (const float*)d_in[10];
  const float* g_prot  = (const float*)d_in[11];
  const float* be_prot = (const float*)d_in[12];
  const int* ei_dd_src = (const int*)d_in[13];
  const int* ei_dd_dst = (const int*)d_in[14];
  const int* ei_dp_src = (const int*)d_in[15];
  const int* ei_dp_dst = (const int*)d_in[16];
  const int* ei_pd_src = (const int*)d_in[17];
  const int* ei_pd_dst = (const int*)d_in[18];
  const int E_DD = in_sizes[13];
  const int E_DP = in_sizes[15];
  const int E_PD = in_sizes[17];

  float* out = (float*)d_out;

  // workspace layout (floats)
  float* ws   = (float*)d_ws;
  float* Y    = ws;                                   // 100000*128 (reused)
  float* S_dd = Y    + (size_t)N_DRUG * DDIM;         // 100000*128
  float* S_pd = S_dd + (size_t)N_DRUG * DDIM;         // 100000*128
  float* S_dp = S_pd + (size_t)N_DRUG * DDIM;         // 50000*128
  float* c_dd = S_dp + (size_t)N_PROT * DDIM;         // 100000
  float* c_pd = c_dd + N_DRUG;                        // 100000
  float* c_dp = c_pd + N_DRUG;                        // 50000

  // zero all accumulators (one contiguous region), graph-capturable
  size_t accBytes = ((size_t)2 * N_DRUG * DDIM + (size_t)N_PROT * DDIM +
                     (size_t)2 * N_DRUG + (size_t)N_PROT) * sizeof(float);
  hipMemsetAsync(S_dd, 0, accBytes, stream);

  dim3 blk(256);

  // relation dd: Y = relu(x_drug @ Wagg_dd^T); scatter into drug dsts
  gemm_relu_xwt<<<N_DRUG / 16, blk, 0, stream>>>(x_drug, Wagg_dd, Y);
  scatter_add<<<(E_DD + 7) / 8, blk, 0, stream>>>(Y, ei_dd_src, ei_dd_dst,
                                                  S_dd, c_dd, E_DD);
  // relation pd: Y = relu(x_prot @ Wagg_pd^T); scatter into drug dsts
  gemm_relu_xwt<<<N_PROT / 16, blk, 0, stream>>>(x_prot, Wagg_pd, Y);
  scatter_add<<<(E_PD + 7) / 8, blk, 0, stream>>>(Y, ei_pd_src, ei_pd_dst,
                                                  S_pd, c_pd, E_PD);
  // relation dp: Y = relu(x_drug @ Wagg_dp^T); scatter into prot dsts
  gemm_relu_xwt<<<N_DRUG / 16, blk, 0, stream>>>(x_drug, Wagg_dp, Y);
  scatter_add<<<(E_DP + 7) / 8, blk, 0, stream>>>(Y, ei_dp_src, ei_dp_dst,
                                                  S_dp, c_dp, E_DP);

  // agg_drug (into S_dd), agg_prot (into S_dp)
  combine_drug<<<((size_t)N_DRUG * DDIM + 255) / 256, blk, 0, stream>>>(
      S_dd, c_dd, S_pd, c_pd, N_DRUG);
  normalize_rows<<<((size_t)N_PROT * DDIM + 255) / 256, blk, 0, stream>>>(
      S_dp, c_dp, N_PROT);

  // node updates -> packed output [drug rows | prot rows]
  node_update<<<N_DRUG / 16, blk, 0, stream>>>(x_drug, S_dd, W_drug, b_drug,
                                               g_drug, be_drug, out, 0);
  node_update<<<N_PROT / 16, blk, 0, stream>>>(x_prot, S_dp, W_prot, b_prot,
                                               g_prot, be_prot, out,
                                               (size_t)N_DRUG);
}